// GCNClassifier_79585743995605
// MI455X (gfx1250) — compile-verified
//
#include <hip/hip_runtime.h>
#include <cstdint>
#include <cstddef>

#define DFE 128  // feature dim

typedef float v2f __attribute__((ext_vector_type(2)));
typedef float v8f __attribute__((ext_vector_type(8)));

// ---------------------------------------------------------------- degrees ---
__global__ void k_init_deg(float* __restrict__ deg_out, float* __restrict__ deg_in, int n) {
  int i = blockIdx.x * blockDim.x + threadIdx.x;
  if (i < n) { deg_out[i] = 1.0f; deg_in[i] = 1.0f; }  // self-loop contributes 1
}

__global__ void k_edge_deg(const int* __restrict__ src, const int* __restrict__ dst,
                           float* __restrict__ deg_out, float* __restrict__ deg_in, int e) {
  int i = blockIdx.x * blockDim.x + threadIdx.x;
  if (i < e) {
    unsafeAtomicAdd(&deg_out[src[i]], 1.0f);
    unsafeAtomicAdd(&deg_in[dst[i]], 1.0f);
  }
}

__global__ void k_rsqrt_deg(float* __restrict__ deg_out, float* __restrict__ deg_in, int n) {
  int i = blockIdx.x * blockDim.x + threadIdx.x;
  if (i < n) {
    deg_out[i] = rsqrtf(deg_out[i]);  // becomes cs
    deg_in[i]  = rsqrtf(deg_in[i]);   // becomes cd
  }
}

// --------------------------------------------- self-loop seed: agg = h*cs ---
// one wave32 per node, float4 per lane (32*4 = 128 features)
__global__ void k_self_init(const float* __restrict__ h, const float* __restrict__ cs,
                            float* __restrict__ agg, int n) {
  int t = blockIdx.x * blockDim.x + threadIdx.x;
  int node = t >> 5, c4 = t & 31;
  if (node < n) {
    float s = cs[node];
    float4 v = ((const float4*)(h + (size_t)node * DFE))[c4];
    v.x *= s; v.y *= s; v.z *= s; v.w *= s;
    ((float4*)(agg + (size_t)node * DFE))[c4] = v;
  }
}

// ------------------------------------------ edge scatter: agg[dst] += h[src]*cs[src]
// one wave32 per edge: coalesced 512B gather, 4 native f32 atomics per lane
__global__ void k_edge_scatter(const int* __restrict__ src, const int* __restrict__ dst,
                               const float* __restrict__ h, const float* __restrict__ cs,
                               float* __restrict__ agg, int e) {
  int t = blockIdx.x * blockDim.x + threadIdx.x;
  int edge = t >> 5, lane = t & 31;
  if (edge < e) {
    int s = src[edge], d = dst[edge];
    float sc = cs[s];
    float4 v = ((const float4*)(h + (size_t)s * DFE))[lane];
    float* p = agg + (size_t)d * DFE + lane * 4;
    unsafeAtomicAdd(p + 0, v.x * sc);
    unsafeAtomicAdd(p + 1, v.y * sc);
    unsafeAtomicAdd(p + 2, v.z * sc);
    unsafeAtomicAdd(p + 3, v.w * sc);
  }
}

// ------------------------------- WMMA GEMM: out = relu((agg*cd) @ W + b) ----
// block = 128 threads = 4 waves; block tile = 16 rows x 128 cols
// each wave: 2 N-tiles of 16, K loop of 32 steps of V_WMMA_F32_16X16X4_F32
__global__ __launch_bounds__(128) void k_gemm_relu(
    const float* __restrict__ agg, const float* __restrict__ cd,
    const float* __restrict__ W, const float* __restrict__ bias,
    float* __restrict__ out, int nrows) {
  __shared__ float As[16 * 132];  // stride 132 to spread banks
  const int tid = threadIdx.x;
  const int row0 = blockIdx.x * 16;

  // stage + scale A tile: 16 rows x 32 float4
  for (int idx = tid; idx < 16 * 32; idx += 128) {
    int r = idx >> 5, c4 = idx & 31;
    int row = row0 + r;
    float4 v = make_float4(0.f, 0.f, 0.f, 0.f);
    float sc = 0.f;
    if (row < nrows) {
      v = ((const float4*)(agg + (size_t)row * DFE))[c4];
      sc = cd[row];
    }
    float* q = As + r * 132 + c4 * 4;
    q[0] = v.x * sc; q[1] = v.y * sc; q[2] = v.z * sc; q[3] = v.w * sc;
  }
  __syncthreads();

  const int wave = tid >> 5;           // wave32
  const int lane = tid & 31;
  const int half = lane >> 4;          // 0: lanes 0-15, 1: lanes 16-31
  const int l15  = lane & 15;
  const int ntile0 = wave * 2;         // this wave's two N-tiles
  const int colb0 = ntile0 * 16;

  v8f acc0 = {};
  v8f acc1 = {};
  const float* Arow = As + l15 * 132;

  for (int k0 = 0; k0 < DFE; k0 += 4) {
    int ka = k0 + 2 * half;            // A-layout: lane half selects K pair
    v2f a = { Arow[ka], Arow[ka + 1] };
    // B-layout: N across lanes 0-15, K pair split across lane halves
    v2f b0 = { W[(size_t)ka * DFE + colb0 + l15],
               W[(size_t)(ka + 1) * DFE + colb0 + l15] };
    v2f b1 = { W[(size_t)ka * DFE + colb0 + 16 + l15],
               W[(size_t)(ka + 1) * DFE + colb0 + 16 + l15] };
    acc0 = __builtin_amdgcn_wmma_f32_16x16x4_f32(false, a, false, b0,
                                                 (short)0, acc0, false, false);
    acc1 = __builtin_amdgcn_wmma_f32_16x16x4_f32(false, a, false, b1,
                                                 (short)0, acc1, false, false);
  }

  // C/D layout: VGPR v -> row (v + 8*half), col = lane&15 within N-tile
  for (int v = 0; v < 8; ++v) {
    int row = row0 + v + 8 * half;
    if (row < nrows) {
      int col0 = colb0 + l15;
      float x0 = acc0[v] + bias[col0];
      out[(size_t)row * DFE + col0] = x0 > 0.f ? x0 : 0.f;
      int col1 = col0 + 16;
      float x1 = acc1[v] + bias[col1];
      out[(size_t)row * DFE + col1] = x1 > 0.f ? x1 : 0.f;
    }
  }
}

// ---------------------------------------------------------------- pooling ---
__global__ void k_pool_zero(float* __restrict__ sums, float* __restrict__ cnt, int g) {
  int i = blockIdx.x * blockDim.x + threadIdx.x;
  if (i < g * DFE) sums[i] = 0.f;
  if (i < g) cnt[i] = 0.f;
}

__global__ void k_pool_acc(const float* __restrict__ h, const int* __restrict__ gid,
                           float* __restrict__ sums, float* __restrict__ cnt, int n) {
  int t = blockIdx.x * blockDim.x + threadIdx.x;
  int node = t >> 5, lane = t & 31;
  if (node < n) {
    int g = gid[node];
    float4 v = ((const float4*)(h + (size_t)node * DFE))[lane];
    float* p = sums + (size_t)g * DFE + lane * 4;
    unsafeAtomicAdd(p + 0, v.x);
    unsafeAtomicAdd(p + 1, v.y);
    unsafeAtomicAdd(p + 2, v.z);
    unsafeAtomicAdd(p + 3, v.w);
    if (lane == 0) unsafeAtomicAdd(&cnt[g], 1.0f);
  }
}

__global__ void k_pool_final(const float* __restrict__ sums, const float* __restrict__ cnt,
                             float* __restrict__ out, int g) {
  int i = blockIdx.x * blockDim.x + threadIdx.x;
  if (i < g * DFE) {
    float c = cnt[i >> 7];
    c = c > 1.0f ? c : 1.0f;
    out[i] = sums[i] / c;
  }
}

// ----------------------------------------------------------------- launch ---
extern "C" void kernel_launch(void* const* d_in, const int* in_sizes, int n_in,
                              void* d_out, int out_size, void* d_ws, size_t ws_size,
                              hipStream_t stream) {
  const float* x   = (const float*)d_in[0];
  const float* W0  = (const float*)d_in[1];
  const float* b0  = (const float*)d_in[2];
  const float* W1  = (const float*)d_in[3];
  const float* b1  = (const float*)d_in[4];
  const float* W2  = (const float*)d_in[5];
  const float* b2  = (const float*)d_in[6];
  const int*   src = (const int*)d_in[7];
  const int*   dst = (const int*)d_in[8];
  const int*   gid = (const int*)d_in[9];

  const int n = in_sizes[0] / DFE;   // 100000
  const int e = in_sizes[7];         // 1600000
  const int g = out_size / DFE;      // 64

  // workspace layout (floats): hA | hB | agg | cs | cd | psum | pcnt
  float* ws = (float*)d_ws;
  size_t ND = (size_t)n * DFE;
  float* hA   = ws;
  float* hB   = hA + ND;
  float* agg  = hB + ND;
  float* cs   = agg + ND;
  float* cd   = cs + n;
  float* psum = cd + n;
  float* pcnt = psum + (size_t)g * DFE;

  const int TB = 256;
  dim3 blk(TB);

  // degrees -> cs, cd
  k_init_deg<<<dim3((n + TB - 1) / TB), blk, 0, stream>>>(cs, cd, n);
  k_edge_deg<<<dim3((e + TB - 1) / TB), blk, 0, stream>>>(src, dst, cs, cd, e);
  k_rsqrt_deg<<<dim3((n + TB - 1) / TB), blk, 0, stream>>>(cs, cd, n);

  const unsigned nodeWaveBlocks = (unsigned)(((size_t)n * 32 + TB - 1) / TB);
  const unsigned edgeWaveBlocks = (unsigned)(((size_t)e * 32 + TB - 1) / TB);
  const unsigned gemmBlocks     = (unsigned)((n + 15) / 16);

  // layer macro: h_in -> h_out
  auto layer = [&](const float* h_in, const float* W, const float* b, float* h_out) {
    k_self_init<<<dim3(nodeWaveBlocks), blk, 0, stream>>>(h_in, cs, agg, n);
    k_edge_scatter<<<dim3(edgeWaveBlocks), blk, 0, stream>>>(src, dst, h_in, cs, agg, e);
    k_gemm_relu<<<dim3(gemmBlocks), dim3(128), 0, stream>>>(agg, cd, W, b, h_out, n);
  };

  layer(x,  W0, b0, hA);
  layer(hA, W1, b1, hB);
  layer(hB, W2, b2, hA);

  // mean pooling per graph
  k_pool_zero<<<dim3((g * DFE + TB - 1) / TB), blk, 0, stream>>>(psum, pcnt, g);
  k_pool_acc<<<dim3(nodeWaveBlocks), blk, 0, stream>>>(hA, gid, psum, pcnt, n);
  k_pool_final<<<dim3((g * DFE + TB - 1) / TB), blk, 0, stream>>>(psum, pcnt, (float*)d_out, g);
}